// SAFL_42597485641738
// MI455X (gfx1250) — compile-verified
//
#include <hip/hip_runtime.h>
#include <hip/hip_bf16.h>
#include <math.h>

#define DIMD 512
#define NTOK 162
#define PARTP 6
#define MROWS 6144           // 1024 * 6
#define HID   2048           // EXP * DIM

typedef __attribute__((ext_vector_type(2))) float v2f;
typedef __attribute__((ext_vector_type(8))) float v8f;

// -------------------------------------------------------------------------
// Kernel 1: per-batch attention + softmax + pooling + LayerNorm (memory-bound)
// one block (256 threads = 8 waves) per batch element
// -------------------------------------------------------------------------
__global__ __launch_bounds__(256)
void attn_pool_ln(const float* __restrict__ x, const float* __restrict__ pt,
                  const float* __restrict__ ln_g, const float* __restrict__ ln_b,
                  float* __restrict__ ynorm)
{
    __shared__ float pt_lds[PARTP * DIMD];        // 12 KB
    __shared__ float attn[PARTP][NTOK + 2];
    __shared__ float wgt[PARTP][NTOK + 2];
    __shared__ float rsum[PARTP], rsq[PARTP];

    const int tid = threadIdx.x;
    const int b   = blockIdx.x;
    const float* xb = x + (size_t)b * NTOK * DIMD;

    for (int i = tid; i < PARTP * DIMD; i += 256) pt_lds[i] = pt[i];
    if (tid < PARTP) { rsum[tid] = 0.0f; rsq[tid] = 0.0f; }
    __syncthreads();

    // ---- phase 1: attn[p][n] = <part_tokens[p], x[b,n,:]> ; thread n owns row n
    if (tid < NTOK) {
        float acc[PARTP] = {0.f, 0.f, 0.f, 0.f, 0.f, 0.f};
        const float* xr = xb + (size_t)tid * DIMD;
        for (int d = 0; d < DIMD; d += 4) {
            float4 xv = *(const float4*)(xr + d);
            #pragma unroll
            for (int p = 0; p < PARTP; ++p) {
                const float* pp = pt_lds + p * DIMD + d;   // LDS broadcast across lanes
                acc[p] += xv.x * pp[0] + xv.y * pp[1] + xv.z * pp[2] + xv.w * pp[3];
            }
        }
        #pragma unroll
        for (int p = 0; p < PARTP; ++p) attn[p][tid] = acc[p];
    }
    __syncthreads();

    // ---- phase 2: softmax over n per p-row; wave p handles row p (wave32 shuffles)
    const int wave = tid >> 5, lane = tid & 31;
    if (wave < PARTP) {
        float mx = -3.4e38f;
        for (int n = lane; n < NTOK; n += 32) mx = fmaxf(mx, attn[wave][n]);
        #pragma unroll
        for (int off = 16; off > 0; off >>= 1) mx = fmaxf(mx, __shfl_xor(mx, off, 32));
        float sum = 0.0f;
        for (int n = lane; n < NTOK; n += 32) {
            float e = __expf(attn[wave][n] - mx);
            wgt[wave][n] = e;
            sum += e;
        }
        #pragma unroll
        for (int off = 16; off > 0; off >>= 1) sum += __shfl_xor(sum, off, 32);
        float inv = 1.0f / sum;
        for (int n = lane; n < NTOK; n += 32) wgt[wave][n] *= inv;
    }
    __syncthreads();

    // ---- phase 3: y[p][d] = sum_n wgt[p][n]*x[b,n,d]; thread owns d0,d0+1 (coalesced)
    const int d0 = tid * 2;
    float acc[PARTP][2] = {};
    for (int n = 0; n < NTOK; ++n) {
        float2 xv = *(const float2*)(xb + (size_t)n * DIMD + d0);
        #pragma unroll
        for (int p = 0; p < PARTP; ++p) {
            float w = wgt[p][n];
            acc[p][0] += w * xv.x;
            acc[p][1] += w * xv.y;
        }
    }
    const float sc = 0.044194173824159216f;   // 1/sqrt(512)
    #pragma unroll
    for (int p = 0; p < PARTP; ++p) {
        acc[p][0] *= sc; acc[p][1] *= sc;
        atomicAdd(&rsum[p], acc[p][0] + acc[p][1]);                       // ds_add_f32
        atomicAdd(&rsq[p],  acc[p][0] * acc[p][0] + acc[p][1] * acc[p][1]);
    }
    __syncthreads();

    float g0 = ln_g[d0], g1 = ln_g[d0 + 1];
    float e0 = ln_b[d0], e1 = ln_b[d0 + 1];
    #pragma unroll
    for (int p = 0; p < PARTP; ++p) {
        float mu  = rsum[p] * (1.0f / DIMD);
        float var = rsq[p]  * (1.0f / DIMD) - mu * mu;
        float rs  = rsqrtf(var + 1e-5f);
        float2 o;
        o.x = (acc[p][0] - mu) * rs * g0 + e0;
        o.y = (acc[p][1] - mu) * rs * g1 + e1;
        *(float2*)(ynorm + ((size_t)(b * PARTP + p) * DIMD + d0)) = o;
    }
}

// -------------------------------------------------------------------------
// fp32 WMMA GEMM: C[M,N] = act(A[M,K] * B[K,N] + bias[N])
// 128 threads = 4 waves; each wave computes a 32x64 tile:
//   2 M-subtiles (dual A fragments) x 4 N-subtiles -> 8x v8f accumulators.
// Per K-chunk of 4: 2 A-loads (b64) + 8 B-loads (b32) feed 8 WMMAs.
// N and K are compile-time so B-row strides fold into 24-bit immediates.
// Fragment layouts per CDNA5 ISA 7.12.2:
//   A 16x4 : lane<16 -> M=lane, {K=0,K=1}; lane>=16 -> M=lane-16, {K=2,K=3}
//   B 4x16 : lane<16 -> N=lane, {K=0,K=1}; lane>=16 -> N=lane-16, {K=2,K=3}
//   C/D    : vgpr r, lane<16 -> (M=r,   N=lane); lane>=16 -> (M=r+8, N=lane-16)
// -------------------------------------------------------------------------
__device__ __forceinline__ float gelu_exact(float v) {
    return 0.5f * v * (1.0f + erff(v * 0.70710678118654752f));
}

template<int ACT, int N, int K>   // ACT: 1 = GELU(.+bias), 0 = .+bias
__global__ __launch_bounds__(128)
void ffn_gemm(const float* __restrict__ A, const float* __restrict__ B,
              const float* __restrict__ bias, float* __restrict__ C)
{
    const int wave   = threadIdx.x >> 5;
    const int lane   = threadIdx.x & 31;
    const int g      = lane >> 4;      // 0: K pair {0,1}, 1: K pair {2,3}
    const int lm     = lane & 15;      // M index (A) / N index (B,C)
    const int m_base = blockIdx.y * 32;
    const int n_base = blockIdx.x * 256 + wave * 64;

    v8f c00 = {}, c01 = {}, c02 = {}, c03 = {};   // M-subtile 0, N-subtiles 0..3
    v8f c10 = {}, c11 = {}, c12 = {}, c13 = {};   // M-subtile 1

    const float* Arow0 = A + (size_t)(m_base + lm) * K;         // rows m_base+lm
    const float* Arow1 = Arow0 + (size_t)16 * K;                // rows m_base+16+lm
    const float* bp    = B + (size_t)(2 * g) * N + n_base + lm; // advances by 4*N per chunk

    #pragma unroll 2
    for (int k0 = 0; k0 < K; k0 += 4) {
        const int kk = k0 + 2 * g;
        v2f a0 = *(const v2f*)(Arow0 + kk);
        v2f a1 = *(const v2f*)(Arow1 + kk);
        __builtin_prefetch(bp + 4 * (size_t)N, 0, 3);   // next K-chunk of B, keep near
        v2f b0 = { bp[0],  bp[N + 0]  };
        v2f b1 = { bp[16], bp[N + 16] };
        v2f b2 = { bp[32], bp[N + 32] };
        v2f b3 = { bp[48], bp[N + 48] };
        bp += 4 * (size_t)N;
        // pair WMMAs sharing the same B fragment back-to-back
        c00 = __builtin_amdgcn_wmma_f32_16x16x4_f32(false, a0, false, b0, (short)0, c00, false, false);
        c10 = __builtin_amdgcn_wmma_f32_16x16x4_f32(false, a1, false, b0, (short)0, c10, false, false);
        c01 = __builtin_amdgcn_wmma_f32_16x16x4_f32(false, a0, false, b1, (short)0, c01, false, false);
        c11 = __builtin_amdgcn_wmma_f32_16x16x4_f32(false, a1, false, b1, (short)0, c11, false, false);
        c02 = __builtin_amdgcn_wmma_f32_16x16x4_f32(false, a0, false, b2, (short)0, c02, false, false);
        c12 = __builtin_amdgcn_wmma_f32_16x16x4_f32(false, a1, false, b2, (short)0, c12, false, false);
        c03 = __builtin_amdgcn_wmma_f32_16x16x4_f32(false, a0, false, b3, (short)0, c03, false, false);
        c13 = __builtin_amdgcn_wmma_f32_16x16x4_f32(false, a1, false, b3, (short)0, c13, false, false);
    }

    const float bs0 = bias[n_base + lm +  0];
    const float bs1 = bias[n_base + lm + 16];
    const float bs2 = bias[n_base + lm + 32];
    const float bs3 = bias[n_base + lm + 48];

    #pragma unroll
    for (int mi = 0; mi < 2; ++mi) {
        const v8f& t0 = mi ? c10 : c00;
        const v8f& t1 = mi ? c11 : c01;
        const v8f& t2 = mi ? c12 : c02;
        const v8f& t3 = mi ? c13 : c03;
        #pragma unroll
        for (int r = 0; r < 8; ++r) {
            const int m = m_base + 16 * mi + 8 * g + r;
            float* cr = C + (size_t)m * N + n_base + lm;
            float v0 = t0[r] + bs0, v1 = t1[r] + bs1, v2 = t2[r] + bs2, v3 = t3[r] + bs3;
            if (ACT) { v0 = gelu_exact(v0); v1 = gelu_exact(v1); v2 = gelu_exact(v2); v3 = gelu_exact(v3); }
            cr[0] = v0; cr[16] = v1; cr[32] = v2; cr[48] = v3;
        }
    }
}

// -------------------------------------------------------------------------
extern "C" void kernel_launch(void* const* d_in, const int* in_sizes, int n_in,
                              void* d_out, int out_size, void* d_ws, size_t ws_size,
                              hipStream_t stream) {
    const float* x    = (const float*)d_in[0];   // [1024,162,512]
    const float* pt   = (const float*)d_in[1];   // [6,512]
    const float* ln_g = (const float*)d_in[2];   // [512]
    const float* ln_b = (const float*)d_in[3];   // [512]
    const float* W1   = (const float*)d_in[4];   // [512,2048]
    const float* b1   = (const float*)d_in[5];   // [2048]
    const float* W2   = (const float*)d_in[6];   // [2048,512]
    const float* b2   = (const float*)d_in[7];   // [512]
    float* out = (float*)d_out;                  // [1024,6,512]

    float* ynorm = (float*)d_ws;                 // [6144,512]  = 12.6 MB
    float* h     = ynorm + (size_t)MROWS * DIMD; // [6144,2048] = 50.3 MB

    (void)in_sizes; (void)n_in; (void)out_size; (void)ws_size;

    attn_pool_ln<<<1024, 256, 0, stream>>>(x, pt, ln_g, ln_b, ynorm);

    // h = gelu(ynorm @ W1 + b1) : [6144,512] x [512,2048]
    ffn_gemm<1, HID, DIMD><<<dim3(HID / 256, MROWS / 32), 128, 0, stream>>>(ynorm, W1, b1, h);

    // out = h @ W2 + b2 : [6144,2048] x [2048,512]
    ffn_gemm<0, DIMD, HID><<<dim3(DIMD / 256, MROWS / 32), 128, 0, stream>>>(h, W2, b2, out);
}